// ARIMA_59373627900094
// MI455X (gfx1250) — compile-verified
//
#include <hip/hip_runtime.h>
#include <hip/hip_bf16.h>
#include <math.h>

typedef __attribute__((ext_vector_type(2))) float v2f;
typedef __attribute__((ext_vector_type(8))) float v8f;

#if defined(__HIP_DEVICE_COMPILE__) && __has_builtin(__builtin_amdgcn_tanhf)
#define FAST_TANH(x) __builtin_amdgcn_tanhf(x)
#else
#define FAST_TANH(x) tanhf(x)
#endif

namespace {
constexpr int NW       = 4;          // waves per block
constexpr int NTHREADS = NW * 32;
constexpr int NB       = 1024;       // batch
constexpr int NC       = 16;         // channels
constexpr int QP1      = 97;         // Q+1
constexpr int HID      = 100;        // hidden
constexpr int NSTEPS   = 100;
constexpr int NT       = 7;          // 7 n-tiles of 16 -> 112 padded outputs
constexpr int WSTRIDE  = 100;        // weight LDS row stride (stride/4 odd -> no bank conflicts)
constexpr int XSTRIDE  = 100;        // state tile stride
constexpr int HSTRIDE  = 116;        // hidden tile stride (29 odd -> no bank conflicts)
// LDS layout (in floats)
constexpr int W1OFF   = 0;                 // [112][100]
constexpr int W2OFF   = 11200;             // [112][100]
constexpr int W3OFF   = 22400;             // [112][100]
constexpr int B1OFF   = 33600;             // [112]
constexpr int B2OFF   = 33712;
constexpr int B3OFF   = 33824;
constexpr int WAVEOFF = 33936;
constexpr int XOFF    = 0;                 // [16][100] per wave
constexpr int H1OFF   = 1600;              // [16][116]
constexpr int H2OFF   = 3456;              // [16][116]
constexpr int WAVESZ  = 5312;
constexpr int LDS_FLOATS = WAVEOFF + NW * WAVESZ;   // 55184 floats = 220,736 B
constexpr int M_TOTAL = NB * NC;                    // 16384 rows
constexpr int NBLOCKS = M_TOTAL / (NW * 16);        // 256
}

// One 16-row x 112-col tile via 7 accumulators; A fragment hoisted per K-step.
__device__ __forceinline__ void gemm_tiles(const float* A, int strideA, int ktiles,
                                           const float* W, const float* bias,
                                           int col, int half2, v8f acc[NT])
{
#pragma unroll
  for (int t = 0; t < NT; ++t) {
    float bb = bias[col + t * 16];
#pragma unroll
    for (int j = 0; j < 8; ++j) acc[t][j] = bb;   // bias broadcast: same n for all 8
  }
#pragma unroll 1
  for (int k = 0; k < ktiles; ++k) {
    const int kk = k * 4 + half2;                 // f32 A/B fragment K offset per half-wave
    v2f a = *(const v2f*)(A + col * strideA + kk);
#pragma unroll
    for (int t = 0; t < NT; ++t) {
      v2f b = *(const v2f*)(W + (col + t * 16) * WSTRIDE + kk);
      acc[t] = __builtin_amdgcn_wmma_f32_16x16x4_f32(
          /*neg_a=*/false, a, /*neg_b=*/false, b,
          /*c_mod=*/(short)0, acc[t], /*reuse_a=*/false, /*reuse_b=*/false);
    }
  }
}

__global__ __launch_bounds__(NTHREADS, 1)
void arima_flow_kernel(const float* __restrict__ series, const float* __restrict__ rand_err,
                       const float* __restrict__ gW1, const float* __restrict__ gb1,
                       const float* __restrict__ gW2, const float* __restrict__ gb2,
                       const float* __restrict__ gW3, const float* __restrict__ gb3,
                       float* __restrict__ out)
{
  extern __shared__ float lds[];
  const int tid     = threadIdx.x;
  const int wave    = tid >> 5;
  const int lane    = tid & 31;
  const int col     = lane & 15;      // N (and M for A-frag) index within tile
  const int halfsel = lane >> 4;      // 0: lanes 0-15, 1: lanes 16-31
  const int half2   = halfsel * 2;

  // --- Stage weights/biases into LDS, zero-padded to 112 x 100 ---
  for (int i = tid; i < 112 * WSTRIDE; i += NTHREADS) {
    const int n = i / WSTRIDE, k = i % WSTRIDE;
    lds[W1OFF + i] = (n < HID && k < 98)  ? gW1[n * 98 + k]  : 0.0f;
    lds[W2OFF + i] = (n < HID && k < HID) ? gW2[n * HID + k] : 0.0f;
    lds[W3OFF + i] = (n < QP1 && k < HID) ? gW3[n * HID + k] : 0.0f;
  }
  for (int i = tid; i < 112; i += NTHREADS) {
    lds[B1OFF + i] = (i < HID) ? gb1[i] : 0.0f;
    lds[B2OFF + i] = (i < HID) ? gb2[i] : 0.0f;
    lds[B3OFF + i] = (i < QP1) ? gb3[i] : 0.0f;
  }
  float* Xb  = lds + WAVEOFF + wave * WAVESZ + XOFF;
  float* H1b = lds + WAVEOFF + wave * WAVESZ + H1OFF;
  float* H2b = lds + WAVEOFF + wave * WAVESZ + H2OFF;
  for (int i = lane; i < 16 * XSTRIDE; i += 32) Xb[i] = 0.0f;  // zero pad cols 98,99 once
  __syncthreads();

  const int row0 = blockIdx.x * (NW * 16) + wave * 16;

  // --- Load noise = cat(series, rand_error) into D-layout register fragments ---
  float xf[NT][8], nf[NT][8];
#pragma unroll
  for (int t = 0; t < NT; ++t) {
    const int n = col + t * 16;
#pragma unroll
    for (int j = 0; j < 8; ++j) {
      const int m = j + halfsel * 8;
      const int r = row0 + m;
      const int b = r >> 4, c = r & 15;
      float v = 0.0f;
      if (n < 96)       v = series[(b * 96 + n) * 16 + c];
      else if (n == 96) v = rand_err[b * 16 + c];
      nf[t][j] = v;
      xf[t][j] = v;       // x starts at noise
    }
  }

  const float dt = 1.0f / NSTEPS;
#pragma unroll 1
  for (int s = 0; s < NSTEPS; ++s) {
    const float tval = (float)s * dt;
    // x fragments -> X tile in LDS (cols 0..96); col 97 = t; 98,99 stay zero
#pragma unroll
    for (int t = 0; t < NT; ++t) {
      const int n = col + t * 16;
      if (n < QP1) {
#pragma unroll
        for (int j = 0; j < 8; ++j)
          Xb[(j + halfsel * 8) * XSTRIDE + n] = xf[t][j];
      }
    }
    if (lane < 16) Xb[lane * XSTRIDE + 97] = tval;

    v8f acc[NT];
    // layer 1: (16 x 98) @ (98 x 100)^T, tanh
    gemm_tiles(Xb, XSTRIDE, 25, lds + W1OFF, lds + B1OFF, col, half2, acc);
#pragma unroll
    for (int t = 0; t < NT; ++t) {
      const int n = col + t * 16;
#pragma unroll
      for (int j = 0; j < 8; ++j)
        H1b[(j + halfsel * 8) * HSTRIDE + n] = FAST_TANH(acc[t][j]);
    }
    // layer 2: (16 x 100) @ (100 x 100)^T, tanh
    gemm_tiles(H1b, HSTRIDE, 25, lds + W2OFF, lds + B2OFF, col, half2, acc);
#pragma unroll
    for (int t = 0; t < NT; ++t) {
      const int n = col + t * 16;
#pragma unroll
      for (int j = 0; j < 8; ++j)
        H2b[(j + halfsel * 8) * HSTRIDE + n] = FAST_TANH(acc[t][j]);
    }
    // layer 3: (16 x 100) @ (100 x 97)^T, then Euler update in registers
    gemm_tiles(H2b, HSTRIDE, 25, lds + W3OFF, lds + B3OFF, col, half2, acc);
#pragma unroll
    for (int t = 0; t < NT; ++t)
#pragma unroll
      for (int j = 0; j < 8; ++j)
        xf[t][j] += (acc[t][j] - nf[t][j]) * dt;   // n>=97 lanes hold junk, never stored
  }

  // --- Write final x (B, 97, 16) ---
#pragma unroll
  for (int t = 0; t < NT; ++t) {
    const int n = col + t * 16;
    if (n < QP1) {
#pragma unroll
      for (int j = 0; j < 8; ++j) {
        const int m = j + halfsel * 8;
        const int r = row0 + m;
        const int b = r >> 4, c = r & 15;
        out[(b * QP1 + n) * 16 + c] = xf[t][j];
      }
    }
  }
}

extern "C" void kernel_launch(void* const* d_in, const int* in_sizes, int n_in,
                              void* d_out, int out_size, void* d_ws, size_t ws_size,
                              hipStream_t stream) {
  (void)in_sizes; (void)n_in; (void)out_size; (void)d_ws; (void)ws_size;
  const float* series   = (const float*)d_in[0];
  const float* rand_err = (const float*)d_in[1];
  const float* W1       = (const float*)d_in[2];
  const float* b1       = (const float*)d_in[3];
  const float* W2       = (const float*)d_in[4];
  const float* b2       = (const float*)d_in[5];
  const float* W3       = (const float*)d_in[6];
  const float* b3       = (const float*)d_in[7];
  float* out            = (float*)d_out;

  const size_t shmem = (size_t)LDS_FLOATS * sizeof(float);  // ~220.7 KB < 320 KB/WGP
  arima_flow_kernel<<<NBLOCKS, NTHREADS, shmem, stream>>>(
      series, rand_err, W1, b1, W2, b2, W3, b3, out);
}